// DilateAttention_8881992368732
// MI455X (gfx1250) — compile-verified
//
#include <hip/hip_runtime.h>
#include <cstdint>
#include <cstddef>

#define HEAD_DIM 32
#define HEADS    12
#define HW       56
#define NPIX     (HW * HW)        /* 3136 */
#define TH       4                /* compute rows per workgroup   */
#define TILES_Y  (HW / TH)        /* 14                           */
#define HROWS    (TH + 4)         /* halo rows (dil=2 -> +-2)     */
#define PITCH    64               /* LDS row pitch (floats)       */
#define BLOCK    (TH * HW)        /* 224 threads = 7 waves (w32)  */
#define SCALE    0.17677669529663687f /* 32^-0.5 */

typedef float v4f __attribute__((ext_vector_type(4)));   // native vec for builtins

// ---- CDNA5 async global->LDS path (guarded; falls back to plain loads) ----
#if defined(__has_builtin)
#  if __has_builtin(__builtin_amdgcn_global_load_async_to_lds_b128)
#    define USE_ASYNC_LDS 1
#    if __has_builtin(__builtin_amdgcn_s_wait_asynccnt)
#      define WAIT_ASYNCCNT0() __builtin_amdgcn_s_wait_asynccnt(0)
#    else
#      define WAIT_ASYNCCNT0() asm volatile("s_wait_asynccnt 0" ::: "memory")
#    endif
#  endif
#endif

#ifdef USE_ASYNC_LDS
typedef int v4i __attribute__((vector_size(16)));
typedef __attribute__((address_space(1))) v4i* gptr_v4i;
typedef __attribute__((address_space(3))) v4i* lptr_v4i;

__device__ __forceinline__ void async_copy16(const float* g, float* l) {
  __builtin_amdgcn_global_load_async_to_lds_b128(
      (gptr_v4i)(g), (lptr_v4i)(l), 0, 0);
}
#endif

__global__ __launch_bounds__(BLOCK)
void dilate_attn_kernel(const float* __restrict__ qg_all,
                        const float* __restrict__ kg_all,
                        const float* __restrict__ vg_all,
                        float* __restrict__ out) {
  // LDS: [head_dim][halo_row][pitch], zero-padded halo; 64 KB each = 128 KB/WG
  __shared__ __align__(16) float kbuf[HEAD_DIM * HROWS * PITCH];
  __shared__ __align__(16) float vbuf[HEAD_DIM * HROWS * PITCH];

  const int tid  = threadIdx.x;
  const int tile = blockIdx.x % TILES_Y;
  const int bh   = blockIdx.x / TILES_Y;        // b*12 + head
  const int y0   = tile * TH;

  // channel plane base: c = head*32+hd, and b*384 + head*32 = bh*32
  const size_t cbase = (size_t)bh * HEAD_DIM * NPIX;
  const float* qg = qg_all + cbase;
  const float* kg = kg_all + cbase;
  const float* vg = vg_all + cbase;

  // ---- fill interior: 32ch * 8rows * 14 float4-quads; 224 threads cover
  // exactly 2 channels per sweep -> 16 iterations, no div/mod in the loop ----
  {
    const int fch0 = tid / (HROWS * 14);   // 0 or 1 (112 units per channel)
    const int fr   = tid % (HROWS * 14);
    const int fyy  = fr / 14;              // 0..7
    const int fqd  = fr % 14;              // 0..13
    const int fy   = y0 - 2 + fyy;
    const bool valid = (unsigned)fy < (unsigned)HW;

    float* lk = &kbuf[(fch0 * HROWS + fyy) * PITCH + 4 + fqd * 4];
    float* lv = &vbuf[(fch0 * HROWS + fyy) * PITCH + 4 + fqd * 4];
    const size_t g0 = (size_t)fch0 * NPIX +
                      (size_t)(valid ? fy : 0) * HW + fqd * 4;
    const float* gk = kg + g0;
    const float* gv = vg + g0;

    if (valid) {
#pragma unroll 4
      for (int ch = fch0; ch < HEAD_DIM; ch += 2) {
#ifdef USE_ASYNC_LDS
        async_copy16(gk, lk);
        async_copy16(gv, lv);
#else
        *(v4f*)lk = *(const v4f*)gk;
        *(v4f*)lv = *(const v4f*)gv;
#endif
        gk += 2 * NPIX;
        gv += 2 * NPIX;
        lk += 2 * HROWS * PITCH;
        lv += 2 * HROWS * PITCH;
      }
    } else {
      const v4f z = {0.f, 0.f, 0.f, 0.f};
#pragma unroll 4
      for (int ch = fch0; ch < HEAD_DIM; ch += 2) {
        *(v4f*)lk = z;
        *(v4f*)lv = z;
        lk += 2 * HROWS * PITCH;
        lv += 2 * HROWS * PITCH;
      }
    }
  }

  // ---- zero the horizontal pad columns xx in {2,3,60,61} (x = -2,-1,56,57)
  // 32ch * 8rows * 4cols = 1024 cells; all strides are powers of two ----
  for (int u = tid; u < HEAD_DIM * HROWS * 4; u += BLOCK) {
    const int ch = u >> 5;          // / (HROWS*4)
    const int r  = u & 31;
    const int yy = r >> 2;
    const int e  = r & 3;
    const int xx = (e < 2) ? (2 + e) : (58 + e);   // 2,3,60,61
    const int idx = (ch * HROWS + yy) * PITCH + xx;
    kbuf[idx] = 0.f;
    vbuf[idx] = 0.f;
  }

  // ---- per-pixel q vector into VGPRs (coalesced, read-once -> nontemporal) ----
  const int ly = tid / HW;
  const int x  = tid % HW;
  const int y  = y0 + ly;
  float qreg[HEAD_DIM];
#pragma unroll
  for (int hd = 0; hd < HEAD_DIM; ++hd)
    qreg[hd] = __builtin_nontemporal_load(
        qg + (size_t)hd * NPIX + (size_t)y * HW + x);

#ifdef USE_ASYNC_LDS
  WAIT_ASYNCCNT0();
#endif
  __syncthreads();

  // ---- scores: 9 dilated neighbors, dy=2(i-1), dx=2(j-1) ----
  float s[9];
#pragma unroll
  for (int i = 0; i < 3; ++i) {
#pragma unroll
    for (int j = 0; j < 3; ++j) {
      const int yy = ly + 2 * i;       // ly + 2 + 2*(i-1)
      const int xx = x + 2 + 2 * j;    // x  + 4 + 2*(j-1)
      float acc = 0.f;
#pragma unroll
      for (int hd = 0; hd < HEAD_DIM; ++hd)
        acc = fmaf(qreg[hd], kbuf[(hd * HROWS + yy) * PITCH + xx], acc);
      s[i * 3 + j] = acc * SCALE;
    }
  }

  // ---- softmax over 9 (zero-padded neighbors legitimately score 0) ----
  float m = s[0];
#pragma unroll
  for (int j = 1; j < 9; ++j) m = fmaxf(m, s[j]);
  float w[9];
  float sum = 0.f;
#pragma unroll
  for (int j = 0; j < 9; ++j) { w[j] = __expf(s[j] - m); sum += w[j]; }
  const float inv = 1.f / sum;
#pragma unroll
  for (int j = 0; j < 9; ++j) w[j] *= inv;

  // ---- weighted sum of V neighbors ----
  float acc[HEAD_DIM];
#pragma unroll
  for (int hd = 0; hd < HEAD_DIM; ++hd) acc[hd] = 0.f;
#pragma unroll
  for (int i = 0; i < 3; ++i) {
#pragma unroll
    for (int j = 0; j < 3; ++j) {
      const int yy = ly + 2 * i;
      const int xx = x + 2 + 2 * j;
      const float wj = w[i * 3 + j];
#pragma unroll
      for (int hd = 0; hd < HEAD_DIM; ++hd)
        acc[hd] = fmaf(wj, vbuf[(hd * HROWS + yy) * PITCH + xx], acc[hd]);
    }
  }

  // ---- store: out[b, y, x, head*32+hd], contiguous 128B per pixel,
  // write-once -> nontemporal so L2 stays dedicated to K/V reuse ----
  const int b    = bh / HEADS;
  const int head = bh % HEADS;
  float* op = out + ((((size_t)b * HW + y) * HW + x) * (HEADS * HEAD_DIM)
                     + head * HEAD_DIM);
#pragma unroll
  for (int hd = 0; hd < HEAD_DIM; hd += 4) {
    v4f r;
    r.x = acc[hd + 0]; r.y = acc[hd + 1]; r.z = acc[hd + 2]; r.w = acc[hd + 3];
    __builtin_nontemporal_store(r, (v4f*)(op + hd));
  }
}

extern "C" void kernel_launch(void* const* d_in, const int* in_sizes, int n_in,
                              void* d_out, int out_size, void* d_ws, size_t ws_size,
                              hipStream_t stream) {
  (void)n_in; (void)out_size; (void)d_ws; (void)ws_size;
  const float* q = (const float*)d_in[0];
  const float* k = (const float*)d_in[1];
  const float* v = (const float*)d_in[2];
  float* out = (float*)d_out;
  const int B = in_sizes[0] / (HEADS * HEAD_DIM * NPIX);   // = 8
  dim3 grid(B * HEADS * TILES_Y);
  dilate_attn_kernel<<<grid, BLOCK, 0, stream>>>(q, k, v, out);
}